// FilterBank_79001628442986
// MI455X (gfx1250) — compile-verified
//
#include <hip/hip_runtime.h>
#include <hip/hip_bf16.h>

// ---------------------------------------------------------------------------
// FilterBank: out[b,t,n,c] = sum_f data[b,c, t*512 + f - 512] * sin(pi*f/1024)
//                                  * filters[f,n]
// 32 independent GEMMs (one per (b,c)): M=513, K=1024, N=1024, fp32,
// V_WMMA_F32_16X16X4_F32 with 32x64 wave tiles and double-buffered LDS.
// ---------------------------------------------------------------------------

typedef float v2f   __attribute__((ext_vector_type(2)));
typedef float v8f   __attribute__((ext_vector_type(8)));
typedef float fvec4 __attribute__((ext_vector_type(4)));

#define FL        1024
#define HOP       512
#define S_LEN     262144
#define T_FRAMES  513
#define N_FILT    1024
#define K_DIM     1024
#define B_BATCH   16
#define C_CH      2

#define BM   128     // output rows (t) per workgroup
#define BN   128     // output cols (n) per workgroup
#define BK   32      // K-block staged in LDS
#define LDA  33      // padded LDS leading dim for A
#define LDB  132     // padded LDS leading dim for B (two-row frag loads fit
                     // in ds_load_2addr's 255-dword offset field)

#define PI_F 3.14159265358979323846f

__global__ __launch_bounds__(256)
void filterbank_wmma_kernel(const float* __restrict__ data,
                            const float* __restrict__ filters,
                            float* __restrict__ out)
{
    __shared__ float As[2][BM * LDA];   // 2 x 16896 B
    __shared__ float Bs[2][BK * LDB];   // 2 x 16896 B
    __shared__ float win[FL];           // 4096 B

    const int tid  = threadIdx.x;
    const int lane = tid & 31;
    const int wave = tid >> 5;       // 0..7
    const int wm   = wave >> 1;      // 0..3 : wave row in 4x2 wave grid
    const int wn   = wave & 1;       // 0..1 : wave col
    const int wmB  = wm * 32;        // wave M base within block tile
    const int wnB  = wn * 64;        // wave N base within block tile

    const int bc = blockIdx.z;       // 0..31  (b*2 + c)
    const int b  = bc >> 1;
    const int c  = bc & 1;
    const int m0 = blockIdx.y * BM;  // frame (t) base
    const int n0 = blockIdx.x * BN;  // filter column base

    const float* __restrict__ dptr = data + (size_t)bc * (size_t)S_LEN;

    const int halfSel = lane >> 4;   // 0 for lanes 0-15, 1 for 16-31
    const int l15     = lane & 15;

    // ---- global -> register staging (block-aligned; a float4 never
    //      straddles the valid sample range since it is 512-aligned) ----
    auto loadA = [&](int kb, fvec4 (&r)[4]) {
        #pragma unroll
        for (int i = 0; i < 4; ++i) {
            const int idx  = i * 256 + tid;
            const int row  = idx >> 3;          // 0..127
            const int col4 = (idx & 7) * 4;     // 0..28
            const int t    = m0 + row;
            const long s   = (long)t * HOP + (kb + col4) - (FL / 2);
            fvec4 v = fvec4{0.f, 0.f, 0.f, 0.f};
            if (t < T_FRAMES && s >= 0 && s < S_LEN)
                v = *(const fvec4*)(dptr + s);
            r[i] = v;
        }
    };
    auto loadB = [&](int kb, fvec4 (&r)[4]) {
        #pragma unroll
        for (int i = 0; i < 4; ++i) {
            const int idx  = i * 256 + tid;
            const int row  = idx >> 5;          // 0..31
            const int col4 = (idx & 31) * 4;    // 0..124
            r[i] = *(const fvec4*)(filters +
                     (size_t)(kb + row) * N_FILT + n0 + col4);
        }
    };
    // ---- register -> LDS (window applied to A here) ----
    auto storeA = [&](int buf, int kb, fvec4 (&r)[4]) {
        #pragma unroll
        for (int i = 0; i < 4; ++i) {
            const int idx  = i * 256 + tid;
            const int row  = idx >> 3;
            const int col4 = (idx & 7) * 4;
            const int f    = kb + col4;
            float* dst = &As[buf][row * LDA + col4];
            dst[0] = r[i][0] * win[f + 0];
            dst[1] = r[i][1] * win[f + 1];
            dst[2] = r[i][2] * win[f + 2];
            dst[3] = r[i][3] * win[f + 3];
        }
    };
    auto storeB = [&](int buf, fvec4 (&r)[4]) {
        #pragma unroll
        for (int i = 0; i < 4; ++i) {
            const int idx  = i * 256 + tid;
            const int row  = idx >> 5;
            const int col4 = (idx & 31) * 4;
            float* dst = &Bs[buf][row * LDB + col4];
            dst[0] = r[i][0];
            dst[1] = r[i][1];
            dst[2] = r[i][2];
            dst[3] = r[i][3];
        }
    };

    // ---- window: win[f] = sqrt(hann)[f] = sin(pi * f / 1024) ----
    for (int i = tid; i < FL; i += 256)
        win[i] = sinf(PI_F * (float)i * (1.0f / (float)FL));

    v8f acc[2][4];
    #pragma unroll
    for (int mi = 0; mi < 2; ++mi)
        #pragma unroll
        for (int ni = 0; ni < 4; ++ni)
            acc[mi][ni] = v8f{0.f, 0.f, 0.f, 0.f, 0.f, 0.f, 0.f, 0.f};

    // ---- prologue: stage K-block 0 into buffer 0 ----
    fvec4 ra[4], rb[4];
    loadA(0, ra);
    loadB(0, rb);
    __syncthreads();            // win[] ready
    storeA(0, 0, ra);
    storeB(0, rb);
    __syncthreads();            // buffer 0 ready

    const int NKB = K_DIM / BK; // 32
    for (int kbi = 0; kbi < NKB; ++kbi) {
        const int  cur  = kbi & 1;
        const bool more = (kbi + 1) < NKB;

        // prefetch next K-block (global -> regs) while we compute
        if (more) {
            loadA((kbi + 1) * BK, ra);
            loadB((kbi + 1) * BK, rb);
        }

        // ---- compute: 8 k-steps x 8 WMMA (2 M-tiles x 4 N-tiles) ----
        const float* __restrict__ Ab = &As[cur][0];
        const float* __restrict__ Bb = &Bs[cur][0];
        #pragma unroll
        for (int k = 0; k < BK; k += 4) {
            // A fragment (16x4): lanes 0-15 hold K=k,k+1 of row M=l15,
            //                    lanes 16-31 hold K=k+2,k+3.
            const int ka = k + (halfSel << 1);

            v2f af[2];
            #pragma unroll
            for (int mi = 0; mi < 2; ++mi) {
                const int row = wmB + mi * 16 + l15;
                af[mi][0] = Ab[row * LDA + ka];
                af[mi][1] = Ab[row * LDA + ka + 1];
            }
            // B fragment (4x16): VGPR0 = rows {k, k+2}, VGPR1 = rows {k+1, k+3}
            v2f bf[4];
            #pragma unroll
            for (int ni = 0; ni < 4; ++ni) {
                const int coln = wnB + ni * 16 + l15;
                bf[ni][0] = Bb[ka * LDB + coln];
                bf[ni][1] = Bb[(ka + 1) * LDB + coln];
            }
            #pragma unroll
            for (int mi = 0; mi < 2; ++mi)
                #pragma unroll
                for (int ni = 0; ni < 4; ++ni)
                    acc[mi][ni] = __builtin_amdgcn_wmma_f32_16x16x4_f32(
                        /*neg_a=*/false, af[mi],
                        /*neg_b=*/false, bf[ni],
                        /*c_mod=*/(short)0, acc[mi][ni],
                        /*reuse_a=*/false, /*reuse_b=*/false);
        }

        // stage next K-block into the other buffer
        if (more) {
            storeA(cur ^ 1, (kbi + 1) * BK, ra);
            storeB(cur ^ 1, rb);
        }
        __syncthreads();
    }

    // ---- store C/D: VGPR g holds M = g + 8*halfSel, N = l15 ----
    // out layout: [B=16][T=513][N=1024][C=2], fp32
    #pragma unroll
    for (int mi = 0; mi < 2; ++mi) {
        #pragma unroll
        for (int ni = 0; ni < 4; ++ni) {
            const int nCol = n0 + wnB + ni * 16 + l15;
            #pragma unroll
            for (int g = 0; g < 8; ++g) {
                const int t = m0 + wmB + mi * 16 + g + halfSel * 8;
                if (t < T_FRAMES) {
                    const size_t o =
                        (((size_t)b * T_FRAMES + t) * N_FILT + nCol) * C_CH + c;
                    out[o] = acc[mi][ni][g];
                }
            }
        }
    }
}

extern "C" void kernel_launch(void* const* d_in, const int* in_sizes, int n_in,
                              void* d_out, int out_size, void* d_ws, size_t ws_size,
                              hipStream_t stream) {
    (void)in_sizes; (void)n_in; (void)out_size; (void)d_ws; (void)ws_size;

    const float* data    = (const float*)d_in[0];   // (16, 2, 262144) f32
    const float* filters = (const float*)d_in[1];   // (1024, 1024)    f32
    float* out           = (float*)d_out;           // (16, 513, 1024, 2) f32

    dim3 grid(N_FILT / BN,                    // 8  N tiles
              (T_FRAMES + BM - 1) / BM,       // 5  M (t) tiles
              B_BATCH * C_CH);                // 32 (b,c) GEMMs
    dim3 block(256);

    filterbank_wmma_kernel<<<grid, block, 0, stream>>>(data, filters, out);
}